// GCN2ATPConv_62723702391588
// MI455X (gfx1250) — compile-verified
//
#include <hip/hip_runtime.h>

typedef __attribute__((ext_vector_type(2))) float v2f;
typedef __attribute__((ext_vector_type(8))) float v8f;

#define CHANNELS 64

// BETA = log(0.5/4 + 1.0) = log(1.125)
__device__ constexpr float BETA_F  = 0.11778303565638346f;
__device__ constexpr float ALPHA_F = 0.1f;

// ---------------------------------------------------------------------------
// 1) deg[v] = 1 (self-loop)
__global__ void init_deg_kernel(float* __restrict__ deg, int n) {
    int v = blockIdx.x * blockDim.x + threadIdx.x;
    if (v < n) deg[v] = 1.0f;
}

// 2) deg[dst_e] += 1 for every edge
__global__ void deg_count_kernel(const int* __restrict__ dst,
                                 float* __restrict__ deg, int E) {
    int e = blockIdx.x * blockDim.x + threadIdx.x;
    if (e < E) atomicAdd(&deg[dst[e]], 1.0f);
}

// 3) deg -> deg^{-1/2} in place (deg >= 1 always, so no clamp branch needed)
__global__ void deg_rsqrt_kernel(float* __restrict__ deg, int n) {
    int v = blockIdx.x * blockDim.x + threadIdx.x;
    if (v < n) deg[v] = rsqrtf(deg[v]);
}

// 4) agg[v][c] = deg^{-1}[v] * x[v][c]   (self-loop contribution, w = dinv^2)
__global__ void agg_init_kernel(const float* __restrict__ x,
                                const float* __restrict__ dinv,
                                float* __restrict__ agg, int total) {
    int idx = blockIdx.x * blockDim.x + threadIdx.x;
    if (idx >= total) return;
    int v = idx >> 6;
    float di = dinv[v];
    agg[idx] = di * di * x[idx];
}

// 5) SpMM scatter: one wave per edge; lane l does channels l and l+32.
__global__ void scatter_kernel(const int* __restrict__ src,
                               const int* __restrict__ dst,
                               const float* __restrict__ x,
                               const float* __restrict__ dinv,
                               float* __restrict__ agg, int E) {
    int gid  = blockIdx.x * blockDim.x + threadIdx.x;
    int e    = gid >> 5;
    int lane = gid & 31;
    if (e >= E) return;
    int s = src[e];
    int d = dst[e];
    float w = dinv[s] * dinv[d];
    const float* xs = x + (size_t)s * CHANNELS;
    float* ad = agg + (size_t)d * CHANNELS;
    atomicAdd(&ad[lane],      w * xs[lane]);
    atomicAdd(&ad[lane + 32], w * xs[lane + 32]);
}

// ---------------------------------------------------------------------------
// 6) Dense GCNII tail with f32 WMMA:
//    h = (1-a)*agg + a*x0 ;  out = (1-b)*h + b*(h @ W1)
//    Seed C = ((1-b)/b)*h, accumulate h@W1 via V_WMMA_F32_16X16X4_F32,
//    scale by b on store.  Block = 256 threads = 8 waves = 128 rows.
__global__ __launch_bounds__(256) void gcn2_gemm_kernel(
        const float* __restrict__ agg, const float* __restrict__ x0,
        const float* __restrict__ W, float* __restrict__ out) {
    __shared__ float sH[8 * 16 * 64];   // 32 KB: 8 per-wave 16x64 h tiles
    __shared__ float sWt[64 * 64];      // 16 KB: W1 transposed [n][k]

    const int t = threadIdx.x;
    const int Rblk = blockIdx.x * 128;  // first row of this block

    // Stage W1 transposed: sWt[n*64+k] = W[k*64+n] (coalesced global reads)
#pragma unroll
    for (int i = 0; i < 16; ++i) {
        int idx = i * 256 + t;          // idx = k*64 + n
        int k = idx >> 6, n = idx & 63;
        sWt[n * 64 + k] = W[idx];
    }
    // Stage h = 0.9*agg + 0.1*x0 for 128 rows (flat layout == row*64+col)
#pragma unroll
    for (int i = 0; i < 32; ++i) {
        int idx = i * 256 + t;          // row*64 + col, row in [0,128)
        size_t g = (size_t)Rblk * 64 + idx;
        sH[idx] = (1.0f - ALPHA_F) * agg[g] + ALPHA_F * x0[g];
    }
    __syncthreads();

    const int wave = t >> 5;
    const int lane = t & 31;
    const int half = lane >> 4;         // 0: lanes 0-15, 1: lanes 16-31
    const int ml   = lane & 15;
    const float* h = &sH[wave * 1024];  // this wave's 16x64 h tile

    // A fragments for all 16 k-steps (reused across 4 column tiles).
    // ISA layout: lane<16 -> {K=4k, K=4k+1}; lane>=16 -> {K=4k+2, K=4k+3}.
    v2f a[16];
#pragma unroll
    for (int k = 0; k < 16; ++k) {
        int off = ml * 64 + 4 * k + 2 * half;
        a[k].x = h[off];
        a[k].y = h[off + 1];
    }

    const float seed = (1.0f - BETA_F) / BETA_F;
#pragma unroll
    for (int nt = 0; nt < 4; ++nt) {
        const int n0 = nt * 16;
        // Seed C with ((1-b)/b)*h ; C layout: VGPR j = row (j + 8*half), col ml
        v8f c;
#pragma unroll
        for (int j = 0; j < 8; ++j)
            c[j] = seed * h[(j + 8 * half) * 64 + n0 + ml];

#pragma unroll
        for (int k = 0; k < 16; ++k) {
            // B fragment: lane holds column n0+ml, rows {4k+2*half, +1}
            int boff = (n0 + ml) * 64 + 4 * k + 2 * half;
            v2f b;
            b.x = sWt[boff];
            b.y = sWt[boff + 1];
            c = __builtin_amdgcn_wmma_f32_16x16x4_f32(
                    false, a[k], false, b, (short)0, c, false, false);
        }

        const int rowBase = Rblk + wave * 16 + 8 * half;
#pragma unroll
        for (int j = 0; j < 8; ++j)
            out[(size_t)(rowBase + j) * 64 + n0 + ml] = BETA_F * c[j];
    }
}

// ---------------------------------------------------------------------------
extern "C" void kernel_launch(void* const* d_in, const int* in_sizes, int n_in,
                              void* d_out, int out_size, void* d_ws, size_t ws_size,
                              hipStream_t stream) {
    const float* x  = (const float*)d_in[0];
    const float* x0 = (const float*)d_in[1];
    const float* W  = (const float*)d_in[2];
    const int*   ei = (const int*)d_in[3];      // edge_index [2, E] flat

    const int N = in_sizes[0] / CHANNELS;       // 65536
    const int E = in_sizes[3] / 2;              // 1048576
    const int* src = ei;
    const int* dst = ei + E;

    float* deg = (float*)d_ws;                  // N floats of scratch
    float* agg = (float*)d_out;                 // accumulate agg in-place

    init_deg_kernel<<<(N + 255) / 256, 256, 0, stream>>>(deg, N);
    deg_count_kernel<<<(E + 255) / 256, 256, 0, stream>>>(dst, deg, E);
    deg_rsqrt_kernel<<<(N + 255) / 256, 256, 0, stream>>>(deg, N);

    const int total = N * CHANNELS;
    agg_init_kernel<<<(total + 255) / 256, 256, 0, stream>>>(x, deg, agg, total);

    // one wave (32 lanes) per edge
    const long long sthreads = (long long)E * 32;
    scatter_kernel<<<(int)((sthreads + 255) / 256), 256, 0, stream>>>(
        src, dst, x, deg, agg, E);

    gcn2_gemm_kernel<<<N / 128, 256, 0, stream>>>(agg, x0, W, (float*)d_out);
}